// NCutsLoss_27023934226975
// MI455X (gfx1250) — compile-verified
//
#include <hip/hip_runtime.h>
#include <hip/hip_bf16.h>

// ---------------------------------------------------------------------------
// NCuts loss for MI455X (gfx1250), wave32.
//
// Shapes (fixed by reference): B=8, K=16, H=W=224, RADIUS=5, WIN=9, PW=PH=232.
// Strategy:
//   * one block per (b, h, 32-pixel tile); 64 threads = 2 waves, 16 px/wave
//   * window contraction expressed as banded GEMM over absolute padded
//     columns -> chained V_WMMA_F32_16X16X4_F32 (all 16 channels x 16 pixels
//     per wave in one 8-VGPR accumulator)
//   * padded_seg rows + per-pixel weights staged in LDS; weight streamed from
//     HBM exactly once (dominant 130 MB of traffic)
//   * per-(b,k) partials accumulated via global_atomic_add_f32 into d_ws
// ---------------------------------------------------------------------------

typedef __attribute__((ext_vector_type(2))) float v2f;
typedef __attribute__((ext_vector_type(8))) float v8f;

#define BB   8
#define KK   16
#define HH   224
#define WW   224
#define PW   232      // padded width/height = 224 + 2*(RADIUS-1)
#define WIN  9
#define TPX  32       // pixels per block
#define WAVES_PER_BLK 2

// LDS padded tile: [16 k][9 rows][44 cols] (40 used; stride 44 => 16B aligned
// rows for b128 staging, even word offsets for b64 A-operand loads)
#define PAD_RSTRIDE 44
#define PAD_KSTRIDE (9 * PAD_RSTRIDE)   // 396
#define PAD_ELEMS   (KK * PAD_KSTRIDE)  // 6336 floats
// LDS weight tile: [32 px][81]
#define WGT_ELEMS   (TPX * 81)          // 2592 floats

__global__ void __launch_bounds__(64)
ncuts_zero_ws(float* ws) {
    ws[threadIdx.x * 4 + 0] = 0.f;   // 64 threads x 4 = 256 floats
    ws[threadIdx.x * 4 + 1] = 0.f;
    ws[threadIdx.x * 4 + 2] = 0.f;
    ws[threadIdx.x * 4 + 3] = 0.f;
}

__global__ void __launch_bounds__(64)
ncuts_main(const float* __restrict__ seg,
           const float* __restrict__ padded,
           const float* __restrict__ weight,
           const float* __restrict__ sum_weight,
           float* __restrict__ wsA,   // [B*K]
           float* __restrict__ wsV) { // [B*K]
    __shared__ __align__(16) float lds_p[PAD_ELEMS];
    __shared__ __align__(16) float lds_w[WGT_ELEMS];

    const int b  = blockIdx.z;
    const int h  = blockIdx.y;
    const int wb = blockIdx.x * TPX;      // first pixel column of block
    const int tid = threadIdx.x;          // 0..63

    // ---- stage weight slab: 32 px * 81 floats, contiguous in HBM ----------
    {
        const float4* src = (const float4*)(weight +
                            (size_t)(((b * HH + h) * WW + wb)) * 81);
        float4* dst = (float4*)lds_w;
        #pragma unroll 2
        for (int i = tid; i < WGT_ELEMS / 4; i += 64) dst[i] = src[i];
    }

    // ---- stage padded_seg window: 16 k * 9 rows * 40 cols ------------------
    {
        float4* dst = (float4*)lds_p;
        for (int i = tid; i < KK * WIN * 10; i += 64) {
            const int row = i / 10;            // 0..143
            const int q   = i - row * 10;      // float4 index within row
            const int k   = row / WIN;
            const int r   = row - k * WIN;
            const float4* src = (const float4*)(padded +
                 (size_t)(((b * KK + k) * PW) + (h + r)) * PW + wb);
            dst[k * (PAD_KSTRIDE / 4) + r * (PAD_RSTRIDE / 4) + q] = src[q];
        }
    }
    __syncthreads();

    const int wave = tid >> 5;            // 0 or 1
    const int lane = tid & 31;
    const int hi   = lane >> 4;           // lane half
    const int l15  = lane & 15;
    const int wt   = wave * 16;           // tile x-offset inside block window

    // A operand: M = channel = lane&15 ; K rows x0..x0+3 with hi-half +2
    const int aBase = l15 * PAD_KSTRIDE + wt + 2 * hi;     // + m*44 + 4c
    // B operand: N = pixel column = lane&15 ; band index n = (4c+2hi+e) - p
    const int pblk  = wt + l15;                            // pixel in block
    const int wBase = pblk * 81;                           // + m*9 + n

    v8f acc = {0.f, 0.f, 0.f, 0.f, 0.f, 0.f, 0.f, 0.f};

    #pragma unroll
    for (int m = 0; m < WIN; ++m) {
        const float* ap = lds_p + aBase + m * PAD_RSTRIDE;
        const float* wp = lds_w + wBase + m * WIN;
        #pragma unroll
        for (int c = 0; c < 6; ++c) {
            // A: two consecutive padded columns, 8B-aligned ds_load_b64
            v2f a = *(const v2f*)(ap + 4 * c);

            // B: banded weights, branchless zero outside [0,8]
            const int n0 = 4 * c + 2 * hi - l15;
            const int n1 = n0 + 1;
            int nc0 = n0 < 0 ? 0 : (n0 > 8 ? 8 : n0);
            int nc1 = n1 < 0 ? 0 : (n1 > 8 ? 8 : n1);
            float b0 = wp[nc0];
            float b1 = wp[nc1];
            b0 = (n0 == nc0) ? b0 : 0.f;
            b1 = (n1 == nc1) ? b1 : 0.f;
            v2f bb = {b0, b1};

            // D[k,p] += A[16x4] * B[4x16]  (chained accumulate)
            acc = __builtin_amdgcn_wmma_f32_16x16x4_f32(
                      false, a, false, bb, (short)0, acc, false, false);
        }
    }

    // ---- epilogue: multiply by seg, reduce 16 lanes/half, atomic into ws ---
    const int pixw = wb + wt + l15;                         // global column
    const float sw = sum_weight[(size_t)(b * HH + h) * WW + pixw];

    #pragma unroll
    for (int r = 0; r < 8; ++r) {
        const int k = r + 8 * hi;                // D vgpr r, hi half = M+8
        const float s = seg[(size_t)(((b * KK + k) * HH) + h) * WW + pixw];
        float pa = acc[r] * s;
        float pv = sw * s;
        #pragma unroll
        for (int off = 1; off < 16; off <<= 1) {
            pa += __shfl_xor(pa, off, 32);
            pv += __shfl_xor(pv, off, 32);
        }
        if (l15 == 0) {
            atomicAdd(&wsA[b * KK + k], pa);
            atomicAdd(&wsV[b * KK + k], pv);
        }
    }
}

__global__ void __launch_bounds__(128)
ncuts_finalize(const float* __restrict__ wsA,
               const float* __restrict__ wsV,
               const int* __restrict__ ncut_k,
               float* __restrict__ out) {
    const int t = threadIdx.x;        // 128 threads: t = b*16 + k
    float r = wsA[t] / wsV[t];
    #pragma unroll
    for (int off = 1; off < 16; off <<= 1) r += __shfl_xor(r, off, 32);
    if ((t & 15) == 0) out[t >> 4] = (float)(*ncut_k) - r;
}

extern "C" void kernel_launch(void* const* d_in, const int* in_sizes, int n_in,
                              void* d_out, int out_size, void* d_ws, size_t ws_size,
                              hipStream_t stream) {
    const float* seg        = (const float*)d_in[0];
    const float* padded_seg = (const float*)d_in[1];
    const float* weight     = (const float*)d_in[2];
    const float* sum_weight = (const float*)d_in[3];
    const int*   ncut_k     = (const int*)d_in[5];   // d_in[4] = radius (unused)

    float* ws  = (float*)d_ws;        // [0..127] assocA, [128..255] assocV
    float* out = (float*)d_out;

    ncuts_zero_ws<<<1, 64, 0, stream>>>(ws);

    dim3 grid(WW / TPX, HH, BB);      // (7, 224, 8)
    ncuts_main<<<grid, 64, 0, stream>>>(seg, padded_seg, weight, sum_weight,
                                        ws, ws + BB * KK);

    ncuts_finalize<<<1, 128, 0, stream>>>(ws, ws + BB * KK, ncut_k, out);
}